// ptPolar_72086731096867
// MI455X (gfx1250) — compile-verified
//
#include <hip/hip_runtime.h>
#include <hip/hip_bf16.h>

// ---------------- problem constants ----------------
#define GX      512
#define GY      512
#define B_TOT   16
#define N_PTS   262144            // 2^18 points per batch
#define P_TOT   (B_TOT * N_PTS)   // 4,194,304 points
#define S_TOT   (B_TOT * GX * GY) // 4,194,304 voxels
#define OUT_TOT (B_TOT * 3 * GX * GY)
#define MAX_PT  64

// ---------------- CDNA5 async global->LDS availability ----------------
#if defined(__gfx1250__) && \
    __has_builtin(__builtin_amdgcn_global_load_async_to_lds_b64) && \
    __has_builtin(__builtin_amdgcn_global_load_async_to_lds_b32) && \
    __has_builtin(__builtin_amdgcn_s_wait_asynccnt)
#define HAVE_ASYNC_LDS 1
#else
#define HAVE_ASYNC_LDS 0
#endif

typedef __attribute__((ext_vector_type(2))) int v2i32;

#if HAVE_ASYNC_LDS
// AS(3) typed pointers, built via integer: generic LDS pointer's low 32 bits
// ARE the LDS offset; AS3 pointers are 32-bit.
typedef __attribute__((address_space(3))) v2i32 lds_v2i32;
typedef __attribute__((address_space(3))) int   lds_i32;
__device__ __forceinline__ lds_v2i32* to_lds_v2(void* p) {
    return (lds_v2i32*)(unsigned)(unsigned long long)p;
}
__device__ __forceinline__ lds_i32* to_lds_i(void* p) {
    return (lds_i32*)(unsigned)(unsigned long long)p;
}
#endif

// ---------------- order-preserving float<->uint (exact min/max via integer atomics) ----
__device__ __forceinline__ unsigned f2ord(float f) {
    unsigned u = __float_as_uint(f);
    return (u & 0x80000000u) ? ~u : (u | 0x80000000u);
}
__device__ __forceinline__ float ord2f(unsigned o) {
    return (o & 0x80000000u) ? __uint_as_float(o & 0x7FFFFFFFu) : __uint_as_float(~o);
}

// Stage one 256-point chunk (xy pair + z) into LDS, async if the async-LDS path exists.
__device__ __forceinline__ void stage_chunk(const float* __restrict__ fea,
                                            const int* __restrict__ xy,
                                            int c, int tid, int* sxyb, float* szb) {
    const int p = (c << 8) + tid;
#if HAVE_ASYNC_LDS
    __builtin_amdgcn_global_load_async_to_lds_b64(
        (v2i32*)(xy + 2 * p), to_lds_v2(sxyb + 2 * tid), 0, 0);
    __builtin_amdgcn_global_load_async_to_lds_b32(
        (int*)(fea + 4 * p + 2), to_lds_i(szb + tid), 0, 0);
#else
    sxyb[2 * tid]     = xy[2 * p];
    sxyb[2 * tid + 1] = xy[2 * p + 1];
    szb[tid]          = fea[4 * p + 2];
    __builtin_prefetch(xy + 2 * (p + 256), 0, 0);   // -> global_prefetch_b8
    __builtin_prefetch(fea + 4 * (p + 256), 0, 0);
#endif
}

__device__ __forceinline__ void wait_chunk(bool more) {
#if HAVE_ASYNC_LDS
    if (more) __builtin_amdgcn_s_wait_asynccnt(2);  // next chunk's 2 ops may stay in flight
    else      __builtin_amdgcn_s_wait_asynccnt(0);
#endif
    (void)more;
}

// ---------------- init workspace ----------------
__global__ void __launch_bounds__(256) k_init(unsigned* __restrict__ cnt,
                                              unsigned long long* __restrict__ zkey,
                                              unsigned* __restrict__ pmin,
                                              unsigned* __restrict__ pmax) {
    const int i = blockIdx.x * 256 + threadIdx.x;
    if (i < S_TOT) {
        cnt[i]  = 0u;
        zkey[i] = 0ull;          // ord part 0 < ord(any finite z)
    }
    if (i < 48) { pmin[i] = 0xFFFFFFFFu; pmax[i] = 0u; }
}

// ---------------- pass 1: counts + fused segment-max(z) / min-index argmax ----------------
// Packed key: {ord(z)[63:32], ~p[31:0]}  ->  u64 atomicMax == (max z, then min index) exactly.
__global__ void __launch_bounds__(256) k_scatter(const float* __restrict__ fea,
                                                 const int* __restrict__ xy,
                                                 unsigned* __restrict__ cnt,
                                                 unsigned long long* __restrict__ zkey) {
    __shared__ int   sxy[2][512];
    __shared__ float sz[2][256];
    const int tid    = threadIdx.x;
    const int nchunk = P_TOT >> 8;
    const int stride = gridDim.x;
    int c = blockIdx.x;
    if (c >= nchunk) return;
    stage_chunk(fea, xy, c, tid, sxy[0], sz[0]);
    int buf = 0;
    while (c < nchunk) {
        const int  nxt  = c + stride;
        const bool more = nxt < nchunk;
        if (more) stage_chunk(fea, xy, nxt, tid, sxy[buf ^ 1], sz[buf ^ 1]);
        wait_chunk(more);
        const int p  = (c << 8) + tid;
        const int xx = sxy[buf][2 * tid];
        const int yy = sxy[buf][2 * tid + 1];
        const float z = sz[buf][tid];
        const unsigned vid = ((unsigned)(p >> 18) << 18) | ((unsigned)xx << 9) | (unsigned)yy;
        const unsigned rank = atomicAdd(&cnt[vid], 1u);
        if (rank < (unsigned)MAX_PT) {
            const unsigned long long key =
                ((unsigned long long)f2ord(z) << 32) | (unsigned long long)(~(unsigned)p);
            atomicMax(&zkey[vid], key);   // GLOBAL_ATOMIC_MAX_U64
        }
        buf ^= 1;
        c = nxt;
    }
}

// ---------------- pass 2: materialize channels + hierarchical plane min/max ----------------
__global__ void __launch_bounds__(256) k_gather(const float* __restrict__ fea,
                                                const unsigned* __restrict__ cnt,
                                                const unsigned long long* __restrict__ zkey,
                                                unsigned* __restrict__ pmin,
                                                unsigned* __restrict__ pmax,
                                                float* __restrict__ out) {
    const int s = blockIdx.x * 256 + threadIdx.x;
    const unsigned c = cnt[s];
    float h0 = 0.0f, r = 0.0f;
    const float cf = (float)(c < (unsigned)MAX_PT ? c : (unsigned)MAX_PT);
    if (c != 0u) {
        const unsigned long long key = zkey[s];
        h0 = ord2f((unsigned)(key >> 32));
        unsigned a = ~(unsigned)key;              // recover min index at max z
        if (a > (unsigned)(P_TOT - 1)) a = (unsigned)(P_TOT - 1);
        r = fea[4u * a + 3u];
    }
    const int b   = s >> 18;         // 2^18 voxels per batch
    const int rem = s & 0x3FFFF;
    out[((b * 3 + 0) << 18) + rem] = h0;
    out[((b * 3 + 1) << 18) + rem] = r;
    out[((b * 3 + 2) << 18) + rem] = cf;

    // wave32 shuffle reduce, then cross-wave LDS reduce, then 6 atomics per block
    unsigned vmn[3] = { f2ord(h0), f2ord(r), f2ord(cf) };
    unsigned vmx[3] = { vmn[0], vmn[1], vmn[2] };
#pragma unroll
    for (int k = 0; k < 3; ++k) {
#pragma unroll
        for (int off = 16; off > 0; off >>= 1) {
            const unsigned omn = __shfl_xor(vmn[k], off, 32);
            const unsigned omx = __shfl_xor(vmx[k], off, 32);
            vmn[k] = min(vmn[k], omn);
            vmx[k] = max(vmx[k], omx);
        }
    }
    __shared__ unsigned smn[8][3], smx[8][3];
    const int wid  = threadIdx.x >> 5;
    const int lane = threadIdx.x & 31;
    if (lane == 0) {
#pragma unroll
        for (int k = 0; k < 3; ++k) { smn[wid][k] = vmn[k]; smx[wid][k] = vmx[k]; }
    }
    __syncthreads();
    if (threadIdx.x == 0) {
#pragma unroll
        for (int k = 0; k < 3; ++k) {
            unsigned mn = smn[0][k], mx = smx[0][k];
#pragma unroll
            for (int w = 1; w < 8; ++w) { mn = min(mn, smn[w][k]); mx = max(mx, smx[w][k]); }
            atomicMin(&pmin[b * 3 + k], mn);
            atomicMax(&pmax[b * 3 + k], mx);
        }
    }
}

// ---------------- pass 3: per-plane min/max normalize ----------------
__global__ void __launch_bounds__(256) k_norm(float* __restrict__ out,
                                              const unsigned* __restrict__ pmin,
                                              const unsigned* __restrict__ pmax) {
    const int t = blockIdx.x * 256 + threadIdx.x;
    if (t >= OUT_TOT) return;
    const int plane = t >> 18;   // uniform per block (256 | 2^18)
    const float mn = ord2f(pmin[plane]);
    const float mx = ord2f(pmax[plane]);
    out[t] = (out[t] - mn) / (mx - mn);
}

extern "C" void kernel_launch(void* const* d_in, const int* in_sizes, int n_in,
                              void* d_out, int out_size, void* d_ws, size_t ws_size,
                              hipStream_t stream) {
    const float* fea = (const float*)d_in[0];   // [16, 262144, 4] f32
    const int*   xy  = (const int*)d_in[1];     // [16, 262144, 2] i32
    float*       out = (float*)d_out;           // [16, 3, 512, 512] f32

    unsigned*           cnt  = (unsigned*)d_ws;                          // S u32
    unsigned long long* zkey = (unsigned long long*)(cnt + S_TOT);       // S u64 (8B aligned)
    unsigned*           pmin = (unsigned*)(zkey + S_TOT);                // 48
    unsigned*           pmax = pmin + 48;                                // 48
    (void)in_sizes; (void)n_in; (void)out_size; (void)ws_size;

    k_init   <<<S_TOT / 256,   256, 0, stream>>>(cnt, zkey, pmin, pmax);
    k_scatter<<<4096,          256, 0, stream>>>(fea, xy, cnt, zkey);
    k_gather <<<S_TOT / 256,   256, 0, stream>>>(fea, cnt, zkey, pmin, pmax, out);
    k_norm   <<<OUT_TOT / 256, 256, 0, stream>>>(out, pmin, pmax);
}